// RollingBufferCache_78520592105598
// MI455X (gfx1250) — compile-verified
//
#include <hip/hip_runtime.h>
#include <stdint.h>

// Rolling-buffer KV cache window gather for MI455X (gfx1250).
// Pure data movement (zero FLOPs): 256 MiB HBM traffic -> ~11.5 us at 23.3 TB/s.
// CDNA5 async global<->LDS DMA path (ASYNCcnt): data never stages through VGPRs;
// one wave32 moves one 512B row per b128 async op (32 lanes x 16B).
// The async IOFFSET is added to BOTH the LDS and global addresses, so the 8
// row-transfers per wave share base registers and differ only in offset:j*512.
// Single code path: per-row guards (uniformly true when W % 64 == 0).

#define BUF_SZ 4096
#define B_     4
#define H_     8
#define S_     512
#define D_     128
#define ROW_BYTES (D_ * 4)          // 512 B per row

#define ROWS_PER_WAVE  8
#define WAVES_PER_BLK  8
#define THREADS_PER_BLK (WAVES_PER_BLK * 32)
#define ROWS_PER_BLK   (ROWS_PER_WAVE * WAVES_PER_BLK)   // 64

template <int OFF>
__device__ __forceinline__ void async_load_b128_off(uint32_t lds_off, uint64_t gaddr) {
    // copies 16B per lane from global address gaddr+OFF to LDS address lds_off+OFF (tracked by ASYNCcnt)
    asm volatile("global_load_async_to_lds_b128 %0, %1, off offset:%2"
                 :: "v"(lds_off), "v"(gaddr), "n"(OFF) : "memory");
}

template <int OFF>
__device__ __forceinline__ void async_store_b128_off(uint64_t gaddr, uint32_t lds_off) {
    // copies 16B per lane from LDS address lds_off+OFF to global address gaddr+OFF
    asm volatile("global_store_async_from_lds_b128 %0, %1, off offset:%2"
                 :: "v"(gaddr), "v"(lds_off), "n"(OFF) : "memory");
}

__device__ __forceinline__ void wait_asynccnt0() {
    asm volatile("s_wait_asynccnt 0x0" ::: "memory");
}

__global__ __launch_bounds__(THREADS_PER_BLK)
void RollingBufferCache_window_kernel(
    const float* __restrict__ k,        // [B,H,S,D]
    const float* __restrict__ v,        // [B,H,S,D]
    const float* __restrict__ k_cache,  // [B,H,BUF,D]
    const float* __restrict__ v_cache,  // [B,H,BUF,D]
    const int*   __restrict__ cur_len_p,
    float*       __restrict__ out,      // [2,B,H,W,D] flat
    int W)                              // min(cur_len, BUF), from out_size
{
    __shared__ __align__(16) unsigned char smem[WAVES_PER_BLK * ROWS_PER_WAVE * ROW_BYTES];

    const int cur_len = *cur_len_p;                    // uniform scalar load
    const int bh      = blockIdx.y;                    // 0 .. B*H-1
    const int tensor  = blockIdx.z;                    // 0 = k, 1 = v

    const float* __restrict__ chunk = tensor ? v       : k;
    const float* __restrict__ cache = tensor ? v_cache : k_cache;

    const int lane = threadIdx.x & 31;
    const int wave = threadIdx.x >> 5;

    // Low 32 bits of the generic LDS pointer == workgroup-relative LDS byte offset.
    const uint32_t lds_base =
        (uint32_t)(uintptr_t)smem + (uint32_t)(wave * ROWS_PER_WAVE * ROW_BYTES)
                                  + (uint32_t)(lane * 16);

    const int w0 = (int)blockIdx.x * ROWS_PER_BLK + wave * ROWS_PER_WAVE;

    const int chunk_start = cur_len - S_;              // first logical time in new chunk
    const int lane_byte   = lane * 16;

    // Per-row global source address, biased by -j*ROW_BYTES so that the shared
    // IOFFSET (offset:j*512) reproduces the true address while also stepping LDS.
    auto src_addr = [&](int j) -> uint64_t {
        const int t = cur_len - W + (w0 + j);          // logical time, >= 0
        const float* src;
        if (t >= chunk_start) {
            src = chunk + ((long long)bh * S_ + (t - chunk_start)) * D_;
        } else {
            src = cache + ((long long)bh * BUF_SZ + (t & (BUF_SZ - 1))) * D_;
        }
        return (uint64_t)(uintptr_t)src + (uint64_t)lane_byte
             - (uint64_t)(j * ROW_BYTES);
    };

    // Destination rows for this (tensor, bh): out[((tensor*BH + bh)*W + w)*D]
    const uint64_t gdst =
        (uint64_t)(uintptr_t)(out + ((long long)(tensor * (B_ * H_) + bh) * W + w0) * D_)
        + (uint64_t)lane_byte;

    uint64_t g[ROWS_PER_WAVE];
#pragma unroll
    for (int j = 0; j < ROWS_PER_WAVE; ++j) g[j] = src_addr(j);

    // ---- Phase 1: 8 async row loads (memory -> LDS), shared LDS base ----
    // Guards are wave-uniform and always true when W % ROWS_PER_BLK == 0.
    if (w0 + 0 < W) async_load_b128_off<0 * ROW_BYTES>(lds_base, g[0]);
    if (w0 + 1 < W) async_load_b128_off<1 * ROW_BYTES>(lds_base, g[1]);
    if (w0 + 2 < W) async_load_b128_off<2 * ROW_BYTES>(lds_base, g[2]);
    if (w0 + 3 < W) async_load_b128_off<3 * ROW_BYTES>(lds_base, g[3]);
    if (w0 + 4 < W) async_load_b128_off<4 * ROW_BYTES>(lds_base, g[4]);
    if (w0 + 5 < W) async_load_b128_off<5 * ROW_BYTES>(lds_base, g[5]);
    if (w0 + 6 < W) async_load_b128_off<6 * ROW_BYTES>(lds_base, g[6]);
    if (w0 + 7 < W) async_load_b128_off<7 * ROW_BYTES>(lds_base, g[7]);

    wait_asynccnt0();   // drain loads before stores consume the staged rows

    // ---- Phase 2: 8 async row stores (LDS -> memory), shared both bases ----
    if (w0 + 0 < W) async_store_b128_off<0 * ROW_BYTES>(gdst, lds_base);
    if (w0 + 1 < W) async_store_b128_off<1 * ROW_BYTES>(gdst, lds_base);
    if (w0 + 2 < W) async_store_b128_off<2 * ROW_BYTES>(gdst, lds_base);
    if (w0 + 3 < W) async_store_b128_off<3 * ROW_BYTES>(gdst, lds_base);
    if (w0 + 4 < W) async_store_b128_off<4 * ROW_BYTES>(gdst, lds_base);
    if (w0 + 5 < W) async_store_b128_off<5 * ROW_BYTES>(gdst, lds_base);
    if (w0 + 6 < W) async_store_b128_off<6 * ROW_BYTES>(gdst, lds_base);
    if (w0 + 7 < W) async_store_b128_off<7 * ROW_BYTES>(gdst, lds_base);

    // S_ENDPGM performs an implicit wait-idle, draining outstanding async stores.
}

extern "C" void kernel_launch(void* const* d_in, const int* in_sizes, int n_in,
                              void* d_out, int out_size, void* d_ws, size_t ws_size,
                              hipStream_t stream) {
    (void)in_sizes; (void)n_in; (void)d_ws; (void)ws_size;

    const float* k       = (const float*)d_in[0];
    const float* v       = (const float*)d_in[1];
    const float* k_cache = (const float*)d_in[2];
    const float* v_cache = (const float*)d_in[3];
    const int*   cur_len = (const int*)d_in[4];
    float*       out     = (float*)d_out;

    // out = [2, B, H, W, D]  ->  W = out_size / (2*B*H*D)   (4096 here)
    const int W = (int)((long long)out_size / (2LL * B_ * H_ * D_));
    const int blocks_x = (W + ROWS_PER_BLK - 1) / ROWS_PER_BLK;

    dim3 grid(blocks_x, B_ * H_, 2);
    RollingBufferCache_window_kernel<<<grid, THREADS_PER_BLK, 0, stream>>>(
        k, v, k_cache, v_cache, cur_len, out, W);
}